// GaussianMixture_16767552324219
// MI455X (gfx1250) — compile-verified
//
#include <hip/hip_runtime.h>
#include <hip/hip_bf16.h>
#include <math.h>

// Problem constants (from reference)
#define BB 8
#define NN 131072
#define KK 4
#define FF 20
#define NITER 5
#define LOG2PI 1.8378770664093453f

// Launch geometry
#define NBLK 64                      // blocks per batch b
#define WPB 8                        // waves per block (256 threads, wave32)
#define NTILES (NN / 16)             // 8192 tiles of 16 points
#define TPW (NTILES / (NBLK * WPB))  // 16 tiles per wave
#define PFDIST 4                     // prefetch distance in tiles

typedef float v2f __attribute__((ext_vector_type(2)));
typedef float v8f __attribute__((ext_vector_type(8)));

static __device__ __forceinline__ v8f wmma4(v2f a, v2f b, v8f c) {
  // D(16x16,f32) = A(16x4,f32) x B(4x16,f32) + C
  return __builtin_amdgcn_wmma_f32_16x16x4_f32(false, a, false, b, (short)0, c,
                                               false, false);
}

// ---------------------------------------------------------------------------
// params layout per b (256 floats):
//   [f*4+k]  f=0..19 :  m[k][f]/v[k][f]         (weight for x)
//   [(20+f)*4+k]     : -0.5/v[k][f]             (weight for x^2)
//   [160+k]          : cinit_k = -0.5*(F*log2pi + sum log v + sum m^2/v)
// partials layout per (b, block) (256 floats):
//   [k*20+f]        : S1  = sum resp * x
//   [80+k*20+f]     : S1b = sum resp^2 * x
//   [160+k*20+f]    : S2b = sum resp^2 * x^2
//   [240+k]         : S0  = sum resp
//   [244+k]         : S0b = sum resp^2
// ---------------------------------------------------------------------------

__global__ __launch_bounds__(64) void gmm_init(const float* __restrict__ means,
                                               const float* __restrict__ covs,
                                               float* __restrict__ params) {
  int t = threadIdx.x;
  if (t < BB * KK) {
    int b = t >> 2, k = t & 3;
    const float* m = means + (size_t)(b * KK + k) * FF;
    const float* v = covs + (size_t)(b * KK + k) * FF;
    float* P = params + b * 256;
    float logdet = 0.f, q = 0.f;
    for (int f = 0; f < FF; ++f) {
      float mf = m[f], vf = v[f];
      P[f * 4 + k] = mf / vf;
      P[(FF + f) * 4 + k] = -0.5f / vf;
      logdet += __logf(vf);
      q += mf * mf / vf;
    }
    P[160 + k] = -0.5f * ((float)FF * LOG2PI + logdet + q);
  }
}

__global__ __launch_bounds__(256) void gmm_estep(
    const float* __restrict__ X, const float* __restrict__ params,
    float* __restrict__ partials, float* __restrict__ resp_out,
    float* __restrict__ lp_out, int write_out) {
  __shared__ float lds_x[WPB][16 * FF];   // x tile per wave
  __shared__ float lds_resp[WPB][64];     // resp tile per wave
  __shared__ float lds_lp[WPB][64];       // log_prob tile per wave
  __shared__ float lds_part[WPB][256];    // per-wave partial sums

  const int tid = threadIdx.x;
  const int w = tid >> 5;
  const int lane = tid & 31;
  const int hi = lane >> 4;   // lane half
  const int n = lane & 15;    // column / point-row index
  const int o = 2 * hi;       // K-offset for A/B operands
  const int b = blockIdx.y;
  const float* P = params + b * 256;

  // Preload B weights (held in VGPRs for the whole N loop).
  // B(4x16) layout: v.x -> row K=o, v.y -> row K=o+1, col N = n (valid n<4).
  v2f Bw[10];
#pragma unroll
  for (int c = 0; c < 10; ++c) {
    int f0 = 4 * c + o;
    float b0 = P[f0 * 4 + (n & 3)];
    float b1 = P[(f0 + 1) * 4 + (n & 3)];
    Bw[c].x = (n < 4) ? b0 : 0.f;
    Bw[c].y = (n < 4) ? b1 : 0.f;
  }
  const float cinit = (n < 4) ? P[160 + n] : 0.f;

  const float* Xb = X + (size_t)b * NN * FF;
  const int stride = NBLK * WPB;  // tile stride between iterations

  v8f accA = {0.f, 0.f, 0.f, 0.f, 0.f, 0.f, 0.f, 0.f};  // resp|resp2 x x[0..15]
  v8f accB = accA;  // resp|resp2 x {x[16..19], 1}
  v8f accC = accA;  // resp2 x x^2[0..15]
  v8f accD = accA;  // resp2 x x^2[16..19]

  const int wave_global = blockIdx.x * WPB + w;

  // Software pipeline: preload tile 0's A operands.
  v2f xa[5];
  {
    const v2f* xp0 =
        (const v2f*)(Xb + (size_t)(wave_global * 16 + n) * FF + o);
#pragma unroll
    for (int c = 0; c < 5; ++c) xa[c] = xp0[2 * c];
  }

#pragma unroll 1
  for (int i = 0; i < TPW; ++i) {
    const int t = wave_global + i * stride;
    const int n0 = t * 16;

    // Issue next tile's global loads now; consumed next iteration.
    const int i_nxt = (i + 1 < TPW) ? (i + 1) : 0;
    const int t_nxt = wave_global + i_nxt * stride;
    v2f xn[5];
    {
      const v2f* xpn = (const v2f*)(Xb + (size_t)(t_nxt * 16 + n) * FF + o);
#pragma unroll
      for (int c = 0; c < 5; ++c) xn[c] = xpn[2 * c];
    }
    // L2 warm-up prefetch a few tiles ahead (global_prefetch_b8).
    {
      int t_pf = wave_global + (i + PFDIST) * stride;
      if (t_pf >= NTILES) t_pf = NTILES - 1;
      __builtin_prefetch(Xb + (size_t)(t_pf * 16 + n) * FF, 0, 3);
    }

    // ---- E-step: log_prob via 10 chained WMMAs (40-long reduction) ----
    v8f d;
#pragma unroll
    for (int r = 0; r < 8; ++r) d[r] = cinit;
#pragma unroll
    for (int c = 0; c < 5; ++c) d = wmma4(xa[c], Bw[c], d);
#pragma unroll
    for (int c = 0; c < 5; ++c) {
      v2f s = xa[c] * xa[c];
      d = wmma4(s, Bw[5 + c], d);
    }

    // resp = (exp(lp)+1e-8) / sum_k : sum over 4-lane group via shfl_xor
    float resp[8], lp[8];
#pragma unroll
    for (int r = 0; r < 8; ++r) {
      lp[r] = d[r];
      float pr = __expf(lp[r]) + 1e-8f;
      float s1 = pr + __shfl_xor(pr, 1, 32);
      float s2 = s1 + __shfl_xor(s1, 2, 32);
      resp[r] = pr / s2;
    }

    // Stage tile in LDS for the M-phase operand rearrangement.
#pragma unroll
    for (int c = 0; c < 5; ++c)
      *(v2f*)&lds_x[w][n * FF + 4 * c + o] = xa[c];
    if (n < 4) {
#pragma unroll
      for (int r = 0; r < 8; ++r)
        lds_resp[w][(r + 8 * hi) * 4 + n] = resp[r];
    }
    if (write_out && n < 4) {
#pragma unroll
      for (int r = 0; r < 8; ++r)
        lds_lp[w][(r + 8 * hi) * 4 + n] = lp[r];
    }
    asm volatile("s_wait_dscnt 0" ::: "memory");

    if (write_out) {  // coalesced tile write: 64 floats per array
      size_t base = ((size_t)b * NN + n0) * KK;
      v2f rv = *(v2f*)&lds_resp[w][2 * lane];
      v2f lv = *(v2f*)&lds_lp[w][2 * lane];
      *(v2f*)&resp_out[base + 2 * lane] = rv;
      *(v2f*)&lp_out[base + 2 * lane] = lv;
    }

    // ---- M-step accumulation: 4 point-groups x 4 WMMAs ----
#pragma unroll
    for (int g = 0; g < 4; ++g) {
      const int p0 = g * 4 + o;  // points p0, p0+1 (K rows o, o+1)
      const int k4 = n & 3;
      float r0 = lds_resp[w][p0 * 4 + k4];
      float r1 = lds_resp[w][(p0 + 1) * 4 + k4];
      v2f a_rr, a_r2;
      a_rr.x = (n < 4) ? r0 : ((n < 8) ? r0 * r0 : 0.f);
      a_rr.y = (n < 4) ? r1 : ((n < 8) ? r1 * r1 : 0.f);
      a_r2.x = (n < 4) ? r0 * r0 : 0.f;
      a_r2.y = (n < 4) ? r1 * r1 : 0.f;
      v2f bx1, bx2, bx2s;
      bx1.x = lds_x[w][p0 * FF + n];
      bx1.y = lds_x[w][(p0 + 1) * FF + n];
      float e0 = lds_x[w][p0 * FF + 16 + (n & 3)];
      float e1 = lds_x[w][(p0 + 1) * FF + 16 + (n & 3)];
      bx2.x = (n < 4) ? e0 : ((n == 4) ? 1.f : 0.f);
      bx2.y = (n < 4) ? e1 : ((n == 4) ? 1.f : 0.f);
      bx2s.x = (n < 4) ? e0 * e0 : ((n == 4) ? 1.f : 0.f);
      bx2s.y = (n < 4) ? e1 * e1 : ((n == 4) ? 1.f : 0.f);
      v2f bx1s = bx1 * bx1;
      accA = wmma4(a_rr, bx1, accA);
      accB = wmma4(a_rr, bx2, accB);
      accC = wmma4(a_r2, bx1s, accC);
      accD = wmma4(a_r2, bx2s, accD);
    }

    // Rotate pipeline registers.
#pragma unroll
    for (int c = 0; c < 5; ++c) xa[c] = xn[c];
  }

  // Dump accumulators (D-layout: lanes<16 hold rows q=r, col f=n). Each slot
  // 0..247 is written by exactly one (lane,r) -> plain stores, deterministic.
  if (hi == 0) {
#pragma unroll
    for (int r = 0; r < 8; ++r) {
      int idx = (r < 4) ? (r * FF + n) : (80 + (r - 4) * FF + n);
      lds_part[w][idx] = accA[r];
    }
    if (n < 5) {
#pragma unroll
      for (int r = 0; r < 8; ++r) {
        int idx;
        if (n < 4)
          idx = (r < 4) ? (r * FF + 16 + n) : (80 + (r - 4) * FF + 16 + n);
        else
          idx = (r < 4) ? (240 + r) : (244 + (r - 4));
        lds_part[w][idx] = accB[r];
      }
    }
#pragma unroll
    for (int r = 0; r < 4; ++r) lds_part[w][160 + r * FF + n] = accC[r];
    if (n < 4) {
#pragma unroll
      for (int r = 0; r < 4; ++r)
        lds_part[w][160 + r * FF + 16 + n] = accD[r];
    }
  }
  __syncthreads();
  if (tid < 248) {  // fixed-order wave reduction -> per-block partials
    float s = 0.f;
#pragma unroll
    for (int ww = 0; ww < WPB; ++ww) s += lds_part[ww][tid];
    partials[((size_t)b * NBLK + blockIdx.x) * 256 + tid] = s;
  }
}

__global__ __launch_bounds__(128) void gmm_mstep(
    const float* __restrict__ partials, float* __restrict__ params,
    float* __restrict__ out_m, float* __restrict__ out_cov,
    float* __restrict__ out_prior, int write_out) {
  __shared__ float S[256];
  const int b = blockIdx.x, t = threadIdx.x;
  for (int idx = t; idx < 256; idx += 128) {
    float s = 0.f;
    for (int blk = 0; blk < NBLK; ++blk)
      s += partials[((size_t)b * NBLK + blk) * 256 + idx];
    S[idx] = s;
  }
  __syncthreads();
  if (write_out) {  // zero full KxFxF cov, diagonals filled below
    for (int idx = t; idx < KK * FF * FF; idx += 128)
      out_cov[(size_t)b * KK * FF * FF + idx] = 0.f;
  }
  __syncthreads();
  if (t < KK) {
    const int k = t;
    const float S0 = S[240 + k], S0b = S[244 + k];
    float* P = params + b * 256;
    float logdet = 0.f, q = 0.f;
    for (int f = 0; f < FF; ++f) {
      float S1 = S[k * FF + f];
      float S1b = S[80 + k * FF + f];
      float S2b = S[160 + k * FF + f];
      float m = S1 / S0;
      float cv = (S2b - 2.f * m * S1b + m * m * S0b) / S0;
      cv = fmaxf(cv, 1e-4f);
      P[f * 4 + k] = m / cv;
      P[(FF + f) * 4 + k] = -0.5f / cv;
      logdet += __logf(cv);
      q += m * m / cv;
      if (write_out) {
        out_m[(size_t)b * KK * FF + k * FF + f] = m;
        out_cov[(size_t)b * KK * FF * FF + k * FF * FF + f * (FF + 1)] = cv;
      }
    }
    P[160 + k] = -0.5f * ((float)FF * LOG2PI + logdet + q);
    if (write_out) out_prior[b * KK + k] = S0;
  }
}

extern "C" void kernel_launch(void* const* d_in, const int* in_sizes, int n_in,
                              void* d_out, int out_size, void* d_ws,
                              size_t ws_size, hipStream_t stream) {
  (void)in_sizes; (void)n_in; (void)out_size; (void)ws_size;
  const float* data = (const float*)d_in[0];
  const float* means = (const float*)d_in[1];
  const float* covar = (const float*)d_in[2];

  float* out = (float*)d_out;
  float* resp_out = out;
  float* lp_out = resp_out + (size_t)BB * NN * KK;
  float* m_out = lp_out + (size_t)BB * NN * KK;
  float* cov_out = m_out + (size_t)BB * KK * FF;
  float* prior_out = cov_out + (size_t)BB * KK * FF * FF;

  float* params = (float*)d_ws;             // 8 * 256 floats
  float* partials = params + BB * 256;      // 8 * NBLK * 256 floats

  gmm_init<<<1, 64, 0, stream>>>(means, covar, params);
  for (int it = 0; it < NITER; ++it) {
    const int wo = (it == NITER - 1) ? 1 : 0;
    gmm_estep<<<dim3(NBLK, BB), 256, 0, stream>>>(data, params, partials,
                                                  resp_out, lp_out, wo);
    gmm_mstep<<<BB, 128, 0, stream>>>(partials, params, m_out, cov_out,
                                      prior_out, wo);
  }
}